// QLayer_65481071399229
// MI455X (gfx1250) — compile-verified
//
#include <hip/hip_runtime.h>

#define TPB 256

// out[b] = ( C1*C2*C3, C0*C1, C0*C1*C2, C0*C1*C2*C3 ),  Cw = cos(x[b][w] + theta[w])
// Derivation: RX gates compose per-wire -> product state; CNOT chain is a
// GF(2)-linear index map; <Z_k> factorizes into products of cos(phi_w).
__global__ __launch_bounds__(TPB) void qlayer_kernel(
    const float4* __restrict__ x, const float* __restrict__ theta,
    float4* __restrict__ out, int n)
{
    __shared__ float4 tile[TPB];
    const int tid = threadIdx.x;
    const int b   = blockIdx.x * TPB + tid;

    // Uniform theta broadcast (scalar loads).
    const float t0 = theta[0];
    const float t1 = theta[1];
    const float t2 = theta[2];
    const float t3 = theta[3];

    // ---- CDNA5 async global->LDS staging (ASYNCcnt path) ----
    // Each lane DMAs its 16B x-row into its LDS slot.
    if (b < n) {
        const float4* gp = x + b;
        // Low 32 bits of a generic pointer into LDS == LDS byte address.
        unsigned lds_off = (unsigned)(size_t)(&tile[tid]);
        asm volatile("global_load_async_to_lds_b128 %0, %1, off"
                     :: "v"(lds_off), "v"(gp)
                     : "memory");
    }
    asm volatile("s_wait_asynccnt 0" ::: "memory");
    __syncthreads();

    if (b < n) {
        const float4 xv = tile[tid];
        const float c0 = __cosf(xv.x + t0);
        const float c1 = __cosf(xv.y + t1);
        const float c2 = __cosf(xv.z + t2);
        const float c3 = __cosf(xv.w + t3);
        const float c12 = c1 * c2;
        float4 o;
        o.x = c12 * c3;   // C1*C2*C3
        o.y = c0 * c1;    // C0*C1
        o.z = c0 * c12;   // C0*C1*C2
        o.w = o.z * c3;   // C0*C1*C2*C3
        out[b] = o;
    }
}

extern "C" void kernel_launch(void* const* d_in, const int* in_sizes, int n_in,
                              void* d_out, int out_size, void* d_ws, size_t ws_size,
                              hipStream_t stream) {
    const float4* x     = (const float4*)d_in[0];   // (BATCH, 4) f32
    const float*  theta = (const float*)d_in[1];    // (4,) f32
    float4*       out   = (float4*)d_out;           // (BATCH, 4) f32

    const int n      = in_sizes[0] / 4;             // batch count
    const int blocks = (n + TPB - 1) / TPB;
    qlayer_kernel<<<blocks, TPB, 0, stream>>>(x, theta, out, n);
}